// PointNetSetAbstraction_67302137528803
// MI455X (gfx1250) — compile-verified
//
#include <hip/hip_runtime.h>

typedef __attribute__((ext_vector_type(2))) float f32x2;
typedef __attribute__((ext_vector_type(8))) float f32x8;

#define NPTS   4096
#define NCTR   1024
#define NB     16
#define NSAMP  32
#define RAD2   0.04f
#define RTOT   524288   // NB*NCTR*NSAMP
#define NCHUNK 512
#define BN_EPS 1e-5f

// ---------------------------------------------------------------------------
// Farthest point sampling: one block per batch, xyz cached in LDS, argmax via
// wave32 shfl_xor butterflies. Matches jnp.argmax tie-break (lowest index).
// Also emits new_xyz directly into d_out[0 .. B*1024*3).
// ---------------------------------------------------------------------------
__global__ __launch_bounds__(256) void fps_kernel(const float* __restrict__ xyz,
                                                  int* __restrict__ fpsIdx,
                                                  float* __restrict__ outXyz) {
  __shared__ float lx[NPTS], ly[NPTS], lz[NPTS];
  __shared__ float rv[8];
  __shared__ int   ri[8];
  __shared__ int   selSlot;
  const int b = blockIdx.x, tid = threadIdx.x;
  const float* base = xyz + (size_t)b * NPTS * 3;
  for (int p = tid; p < NPTS; p += 256) {
    lx[p] = base[p * 3 + 0];
    ly[p] = base[p * 3 + 1];
    lz[p] = base[p * 3 + 2];
  }
  __syncthreads();

  float md[16];
#pragma unroll
  for (int j = 0; j < 16; ++j) md[j] = __builtin_inff();
  int sel = 0;
  for (int t = 0; t < NCTR; ++t) {
    if (tid == 0) {
      fpsIdx[b * NCTR + t] = sel;
      size_t o = ((size_t)b * NCTR + t) * 3;
      outXyz[o + 0] = lx[sel];
      outXyz[o + 1] = ly[sel];
      outXyz[o + 2] = lz[sel];
    }
    const float cx = lx[sel], cy = ly[sel], cz = lz[sel];
    float best = -1.f;
    int bidx = 0;
#pragma unroll
    for (int j = 0; j < 16; ++j) {
      const int p = tid + j * 256;
      const float dx = lx[p] - cx, dy = ly[p] - cy, dz = lz[p] - cz;
      const float d = dx * dx + dy * dy + dz * dz;
      md[j] = fminf(md[j], d);
      if (md[j] > best) { best = md[j]; bidx = p; }   // ascending p => first max
    }
#pragma unroll
    for (int off = 16; off >= 1; off >>= 1) {
      const float ov = __shfl_xor(best, off, 32);
      const int   oi = __shfl_xor(bidx, off, 32);
      if (ov > best || (ov == best && oi < bidx)) { best = ov; bidx = oi; }
    }
    if ((tid & 31) == 0) { rv[tid >> 5] = best; ri[tid >> 5] = bidx; }
    __syncthreads();
    if (tid == 0) {
      float bv = rv[0];
      int bi = ri[0];
      for (int w = 1; w < 8; ++w)
        if (rv[w] > bv || (rv[w] == bv && ri[w] < bi)) { bv = rv[w]; bi = ri[w]; }
      selSlot = bi;
    }
    __syncthreads();
    sel = selSlot;
  }
}

// ---------------------------------------------------------------------------
// Ball query (top-32 smallest distances, stable ties => matches argsort) +
// radius mask + gather of [grouped_xyz | grouped_feat | pad0] into X0 (R x 68).
// One block per center.
// ---------------------------------------------------------------------------
__global__ __launch_bounds__(256) void ballq_gather_kernel(
    const float* __restrict__ xyz, const float* __restrict__ feat,
    const int* __restrict__ fpsIdx, float* __restrict__ X0) {
  __shared__ float dist[NPTS];
  __shared__ float rv[8];
  __shared__ int   ri[8];
  __shared__ int   selI[NSAMP];
  __shared__ float selD[NSAMP];
  const int tid = threadIdx.x;
  const int bm = blockIdx.x;
  const int b = bm >> 10;
  const int ci = fpsIdx[bm];
  const float* base = xyz + (size_t)b * NPTS * 3;
  const float cx = base[ci * 3 + 0], cy = base[ci * 3 + 1], cz = base[ci * 3 + 2];
  for (int p = tid; p < NPTS; p += 256) {
    const float dx = base[p * 3 + 0] - cx, dy = base[p * 3 + 1] - cy,
                dz = base[p * 3 + 2] - cz;
    dist[p] = dx * dx + dy * dy + dz * dz;
  }
  __syncthreads();
  for (int s = 0; s < NSAMP; ++s) {
    float best = __builtin_inff();
    int bidx = 0x7fffffff;
#pragma unroll
    for (int j = 0; j < 16; ++j) {
      const int p = tid + j * 256;
      const float v = dist[p];
      if (v < best) { best = v; bidx = p; }  // ascending p => first min
    }
#pragma unroll
    for (int off = 16; off >= 1; off >>= 1) {
      const float ov = __shfl_xor(best, off, 32);
      const int   oi = __shfl_xor(bidx, off, 32);
      if (ov < best || (ov == best && oi < bidx)) { best = ov; bidx = oi; }
    }
    if ((tid & 31) == 0) { rv[tid >> 5] = best; ri[tid >> 5] = bidx; }
    __syncthreads();
    if (tid == 0) {
      float bv = rv[0];
      int bi = ri[0];
      for (int w = 1; w < 8; ++w)
        if (rv[w] < bv || (rv[w] == bv && ri[w] < bi)) { bv = rv[w]; bi = ri[w]; }
      selI[s] = bi;
      selD[s] = bv;
      dist[bi] = __builtin_inff();
    }
    __syncthreads();
  }
  const int i0 = selI[0];
  const float* fbase = feat + (size_t)b * NPTS * 64;
  for (int e = tid; e < NSAMP * 68; e += 256) {
    const int s = e / 68, c = e % 68;
    const int idx = (selD[s] > RAD2) ? i0 : selI[s];
    float v;
    if (c < 3) {
      const float ctr = (c == 0) ? cx : ((c == 1) ? cy : cz);
      v = base[idx * 3 + c] - ctr;
    } else if (c < 67) {
      v = fbase[(size_t)idx * 64 + (c - 3)];
    } else {
      v = 0.f;  // K-pad column
    }
    X0[((size_t)bm * NSAMP + s) * 68 + c] = v;
  }
}

// ---------------------------------------------------------------------------
// Fused (BN+ReLU on input) + GEMM  Y = act(X) @ W^T  via v_wmma_f32_16x16x4_f32.
// 256 threads = 8 waves; each wave owns a 16-row tile, loops over 16-col tiles.
// Weights live in LDS in a K-pair-interleaved layout so each B fragment is one
// contiguous 8B ds_load_b64 into an even VGPR pair (no marshaling movs, no
// bank conflicts).  LDS index: (c>>1)*(2*COUT) + n*2 + (c&1).
// ---------------------------------------------------------------------------
template <int CIN, int CINP, int COUT, int ACT>
__global__ __launch_bounds__(256) void gemm_bn_wmma(
    const float* __restrict__ X, const float* __restrict__ W,
    const float* __restrict__ scale, const float* __restrict__ shift,
    float* __restrict__ Y) {
  constexpr int KST = CINP / 4;
  constexpr int NT = COUT / 16;
  __shared__ float Wt[CINP * COUT];  // interleaved: [(c>>1)][n][c&1]
  __shared__ float sc[CINP], sh[CINP];
  for (int e = threadIdx.x; e < CINP * COUT; e += 256) {
    const int d = e / CINP, c = e % CINP;
    const float v = (c < CIN) ? W[d * CIN + c] : 0.f;
    Wt[(c >> 1) * (2 * COUT) + d * 2 + (c & 1)] = v;
  }
  if (ACT) {
    for (int e = threadIdx.x; e < CINP; e += 256) {
      sc[e] = (e < CIN) ? scale[e] : 0.f;
      sh[e] = (e < CIN) ? shift[e] : 0.f;
    }
  }
  __syncthreads();

  const int lane = threadIdx.x & 31;
  const int wave = threadIdx.x >> 5;
  const int row0 = (blockIdx.x * 8 + wave) * 16;
  const int mrow = lane & 15;
  const int kh = (lane >> 4) * 2;  // 0 or 2 : K-pair selector per lane-half
  const float* xr = X + (size_t)(row0 + mrow) * CINP;

  float ax[KST], ay[KST];
#pragma unroll
  for (int k = 0; k < KST; ++k) {
    const int c0 = k * 4 + kh;
    f32x2 v = *(const f32x2*)(xr + c0);  // 8B-aligned: stride & c0 even
    float v0 = v[0], v1 = v[1];
    if (ACT) {
      v0 = fmaxf(0.f, fmaf(v0, sc[c0], sh[c0]));
      v1 = fmaxf(0.f, fmaf(v1, sc[c0 + 1], sh[c0 + 1]));
    }
    ax[k] = v0;
    ay[k] = v1;
  }

  const int nlane = lane & 15;
#pragma unroll
  for (int nt = 0; nt < NT; ++nt) {
    const int ncol = nt * 16 + nlane;
    f32x8 acc = {};
#pragma unroll
    for (int k = 0; k < KST; ++k) {
      const int c0 = k * 4 + kh;  // even
      f32x2 a;
      a[0] = ax[k];
      a[1] = ay[k];
      const f32x2 bb =
          *(const f32x2*)&Wt[(c0 >> 1) * (2 * COUT) + ncol * 2];  // one b64
      acc = __builtin_amdgcn_wmma_f32_16x16x4_f32(false, a, false, bb,
                                                  (short)0, acc, false, false);
    }
#pragma unroll
    for (int i = 0; i < 8; ++i) {
      const int Mr = i + (lane >> 4) * 8;
      Y[(size_t)(row0 + Mr) * COUT + ncol] = acc[i];
    }
  }
}

// ---------------------------------------------------------------------------
// Deterministic per-channel sum / sum-of-squares over R rows (2-stage tree).
// ---------------------------------------------------------------------------
__global__ __launch_bounds__(256) void stats_kernel(const float* __restrict__ Y,
                                                    float* __restrict__ part,
                                                    int Cout) {
  const int rowsPer = RTOT / NCHUNK;  // 1024
  const int chunk = blockIdx.x;
  const int c = threadIdx.x % Cout;
  const int ph = threadIdx.x / Cout;
  const int P = 256 / Cout;
  float s = 0.f, ss = 0.f;
  for (int r = ph; r < rowsPer; r += P) {
    const float v = Y[((size_t)chunk * rowsPer + r) * Cout + c];
    s += v;
    ss += v * v;
  }
  __shared__ float sb[256], qb[256];
  sb[threadIdx.x] = s;
  qb[threadIdx.x] = ss;
  __syncthreads();
  if (ph == 0) {
    for (int q = 1; q < P; ++q) {
      s += sb[c + q * Cout];
      ss += qb[c + q * Cout];
    }
    part[(size_t)chunk * Cout + c] = s;
    part[(size_t)NCHUNK * Cout + (size_t)chunk * Cout + c] = ss;
  }
}

__global__ void finalize_kernel(const float* __restrict__ part,
                                const float* __restrict__ gamma,
                                const float* __restrict__ beta,
                                float* __restrict__ scale,
                                float* __restrict__ shift, int Cout) {
  const int c = threadIdx.x;
  if (c >= Cout) return;
  float s = 0.f, ss = 0.f;
  for (int ch = 0; ch < NCHUNK; ++ch) {
    s += part[(size_t)ch * Cout + c];
    ss += part[(size_t)NCHUNK * Cout + (size_t)ch * Cout + c];
  }
  const float inv = 1.f / (float)RTOT;
  const float mean = s * inv;
  float var = ss * inv - mean * mean;
  var = fmaxf(var, 0.f);
  const float scv = gamma[c] * rsqrtf(var + BN_EPS);
  scale[c] = scv;
  shift[c] = beta[c] - mean * scv;
}

// ---------------------------------------------------------------------------
// Final layer BN+ReLU fused with max over the 32 samples.
// ---------------------------------------------------------------------------
__global__ __launch_bounds__(128) void maxpool_kernel(
    const float* __restrict__ Y2, const float* __restrict__ scale,
    const float* __restrict__ shift, float* __restrict__ out) {
  const int d = threadIdx.x;
  const size_t m = blockIdx.x;
  const float scv = scale[d], shv = shift[d];
  float acc = 0.f;  // relu outputs are >= 0
  for (int s = 0; s < NSAMP; ++s) {
    const float v = Y2[(m * NSAMP + s) * 128 + d];
    acc = fmaxf(acc, fmaxf(0.f, fmaf(v, scv, shv)));
  }
  out[m * 128 + d] = acc;
}

extern "C" void kernel_launch(void* const* d_in, const int* in_sizes, int n_in,
                              void* d_out, int out_size, void* d_ws,
                              size_t ws_size, hipStream_t stream) {
  const float* xyz  = (const float*)d_in[0];
  const float* feat = (const float*)d_in[1];
  const float* w0 = (const float*)d_in[2];
  const float* g0 = (const float*)d_in[4];
  const float* be0 = (const float*)d_in[5];
  const float* w1 = (const float*)d_in[6];
  const float* g1 = (const float*)d_in[8];
  const float* be1 = (const float*)d_in[9];
  const float* w2 = (const float*)d_in[10];
  const float* g2 = (const float*)d_in[12];
  const float* be2 = (const float*)d_in[13];

  float* out = (float*)d_out;
  float* outXyz = out;                           // (16,1024,3)
  float* outFeat = out + (size_t)NB * NCTR * 3;  // (16,1024,128)

  const size_t R = RTOT;
  float* ws = (float*)d_ws;
  float* x0   = ws;                    // R*68  (layer0 input; later aliased as Y1)
  float* yB   = x0 + R * 68;           // R*128 (Y0 in first R*64, later Y2)
  float* part = yB + R * 128;          // 512*128*2
  float* scl  = part + (size_t)NCHUNK * 128 * 2;  // 128
  float* shf  = scl + 128;                        // 128
  int* fpsIdx = (int*)(shf + 128);                // 16*1024
  float* y1 = x0;  // alias: x0 dead after layer-0 GEMM

  fps_kernel<<<NB, 256, 0, stream>>>(xyz, fpsIdx, outXyz);
  ballq_gather_kernel<<<NB * NCTR, 256, 0, stream>>>(xyz, feat, fpsIdx, x0);

  // Layer 0: 67 -> 64 (raw input, no activation; bias cancels under BN)
  gemm_bn_wmma<67, 68, 64, 0><<<R / 128, 256, 0, stream>>>(x0, w0, nullptr,
                                                           nullptr, yB);
  stats_kernel<<<NCHUNK, 256, 0, stream>>>(yB, part, 64);
  finalize_kernel<<<1, 128, 0, stream>>>(part, g0, be0, scl, shf, 64);

  // Layer 1: 64 -> 64 (BN0+ReLU fused into A-fragment load)
  gemm_bn_wmma<64, 64, 64, 1><<<R / 128, 256, 0, stream>>>(yB, w1, scl, shf, y1);
  stats_kernel<<<NCHUNK, 256, 0, stream>>>(y1, part, 64);
  finalize_kernel<<<1, 128, 0, stream>>>(part, g1, be1, scl, shf, 64);

  // Layer 2: 64 -> 128 (BN1+ReLU fused)
  gemm_bn_wmma<64, 64, 128, 1><<<R / 128, 256, 0, stream>>>(y1, w2, scl, shf, yB);
  stats_kernel<<<NCHUNK, 256, 0, stream>>>(yB, part, 128);
  finalize_kernel<<<1, 128, 0, stream>>>(part, g2, be2, scl, shf, 128);

  // BN2+ReLU + max over samples
  maxpool_kernel<<<NB * NCTR, 128, 0, stream>>>(yB, scl, shf, outFeat);
}